// DICELoss3D_3186865733987
// MI455X (gfx1250) — compile-verified
//
#include <hip/hip_runtime.h>
#include <hip/hip_bf16.h>

// Bandwidth-bound masked log-softmax NLL reduction for MI455X (gfx1250).
//   result = (1/65536) * sum over all voxels of ( logsumexp_c(v) - v[mask] )
// Streams: 4 contiguous channel planes (16 MiB each per batch) + mask plane,
// all read exactly once -> non-temporal b128 loads, grid-stride accumulation,
// deterministic two-kernel reduction. Block reduction uses V_WMMA_F32_16X16X4_F32
// (ones(16x4) x partials(4x16) accumulated into D) to collapse 256 partials.

typedef float v4f __attribute__((ext_vector_type(4)));
typedef int   v4i __attribute__((ext_vector_type(4)));
typedef float v2f __attribute__((ext_vector_type(2)));
typedef float v8f __attribute__((ext_vector_type(8)));

#define NLL_NBLOCKS  1024
#define NLL_NTHREADS 256

__device__ __forceinline__ float nll_voxel(float v0, float v1, float v2, float v3, int m) {
    float mx  = fmaxf(fmaxf(v0, v1), fmaxf(v2, v3));
    float s   = __expf(v0 - mx) + __expf(v1 - mx) + __expf(v2 - mx) + __expf(v3 - mx);
    float lse = mx + __logf(s);
    float p   = (m == 0) ? v0 : ((m == 1) ? v1 : ((m == 2) ? v2 : v3));
    return lse - p;
}

__global__ __launch_bounds__(NLL_NTHREADS)
void nll_partial_kernel(const float* __restrict__ logits,
                        const int*   __restrict__ mask,
                        float*       __restrict__ block_partials) {
    // Tensor geometry: [B=2, C=4, X=256, Y=256, Z=64]; S = X*Y*Z = 2^22.
    const int S = 1 << 22;
    const int V = (2 * S) >> 2;                 // vec4 voxel groups over all pixels

    float acc = 0.0f;
    int tid    = blockIdx.x * blockDim.x + threadIdx.x;
    int stride = gridDim.x * blockDim.x;

    for (int g = tid; g < V; g += stride) {
        int p = g << 2;                         // voxel index (z-contiguous)
        int b = p >> 22;
        int s = p & (S - 1);
        const float* base = logits + ((size_t)(b * 4) << 22) + s;

        // 4 coalesced channel streams + mask stream, single-use -> NT loads.
        v4f c0 = __builtin_nontemporal_load((const v4f*)(base));
        v4f c1 = __builtin_nontemporal_load((const v4f*)(base + S));
        v4f c2 = __builtin_nontemporal_load((const v4f*)(base + 2 * S));
        v4f c3 = __builtin_nontemporal_load((const v4f*)(base + 3 * S));
        v4i mk = __builtin_nontemporal_load((const v4i*)(mask + p));

        acc += nll_voxel(c0.x, c1.x, c2.x, c3.x, mk.x);
        acc += nll_voxel(c0.y, c1.y, c2.y, c3.y, mk.y);
        acc += nll_voxel(c0.z, c1.z, c2.z, c3.z, mk.z);
        acc += nll_voxel(c0.w, c1.w, c2.w, c3.w, mk.w);
    }

    __shared__ float parts[NLL_NTHREADS];
    parts[threadIdx.x] = acc;
    __syncthreads();

    int lane = threadIdx.x & 31;

#if __has_builtin(__builtin_amdgcn_wmma_f32_16x16x4_f32)
    // Block reduction via WMMA: D = ones(16x4) x B(4x16) + D sums each B column
    // and broadcasts down all 16 rows. Four chunks of 64 partials -> one D whose
    // row 0 (VGPR0, lanes 0..15) holds 16 column totals covering all 256 values.
    // All 8 waves compute redundantly (uniform control flow, EXEC all-ones).
    v2f a = {1.0f, 1.0f};
    v8f d = {0.0f, 0.0f, 0.0f, 0.0f, 0.0f, 0.0f, 0.0f, 0.0f};
#pragma unroll
    for (int chunk = 0; chunk < 4; ++chunk) {
        v2f bm;
        bm.x = parts[chunk * 64 + lane];
        bm.y = parts[chunk * 64 + 32 + lane];
        d = __builtin_amdgcn_wmma_f32_16x16x4_f32(
                /*neg_a=*/false, a, /*neg_b=*/false, bm,
                /*c_mod=*/(short)0, d, /*reuse_a=*/false, /*reuse_b=*/false);
    }
    float r = d[0];                 // lanes 0..15: row M=0; lanes 16..31: identical row M=8
    r += __shfl_xor(r, 1, 32);
    r += __shfl_xor(r, 2, 32);
    r += __shfl_xor(r, 4, 32);
    r += __shfl_xor(r, 8, 32);      // lane 0 now holds the 256-way block sum
#else
    // Fallback: shuffle tree within wave, serial sum of wave leaders.
    float r = acc;
    for (int o = 16; o > 0; o >>= 1) r += __shfl_down(r, o, 32);
    __shared__ float wsum[NLL_NTHREADS / 32];
    if (lane == 0) wsum[threadIdx.x >> 5] = r;
    __syncthreads();
    if (threadIdx.x == 0) {
        r = 0.0f;
        for (int w = 0; w < NLL_NTHREADS / 32; ++w) r += wsum[w];
    }
#endif

    if (threadIdx.x == 0) block_partials[blockIdx.x] = r;
}

__global__ __launch_bounds__(NLL_NTHREADS)
void nll_final_kernel(const float* __restrict__ block_partials,
                      float*       __restrict__ out,
                      int n) {
    float acc = 0.0f;
    for (int i = threadIdx.x; i < n; i += blockDim.x) acc += block_partials[i];

    // Deterministic tree: wave shuffle reduce, leaders -> LDS, thread 0 sums.
    for (int o = 16; o > 0; o >>= 1) acc += __shfl_down(acc, o, 32);
    __shared__ float wsum[NLL_NTHREADS / 32];
    if ((threadIdx.x & 31) == 0) wsum[threadIdx.x >> 5] = acc;
    __syncthreads();
    if (threadIdx.x == 0) {
        float t = 0.0f;
        for (int w = 0; w < NLL_NTHREADS / 32; ++w) t += wsum[w];
        out[0] = t * (1.0f / 65536.0f);   // per-slice mean over X*Y, summed over B*Z
    }
}

extern "C" void kernel_launch(void* const* d_in, const int* in_sizes, int n_in,
                              void* d_out, int out_size, void* d_ws, size_t ws_size,
                              hipStream_t stream) {
    (void)in_sizes; (void)n_in; (void)out_size; (void)ws_size;
    const float* logits = (const float*)d_in[0];
    const int*   mask   = (const int*)d_in[1];   // jax default x64-off -> int32 ids
    float* partials     = (float*)d_ws;          // NLL_NBLOCKS * 4 bytes of scratch
    float* out          = (float*)d_out;

    nll_partial_kernel<<<NLL_NBLOCKS, NLL_NTHREADS, 0, stream>>>(logits, mask, partials);
    nll_final_kernel<<<1, NLL_NTHREADS, 0, stream>>>(partials, out, NLL_NBLOCKS);
}